// ContinuousConvolutionBlock_25434796327480
// MI455X (gfx1250) — compile-verified
//
#include <hip/hip_runtime.h>

// ---------------------------------------------------------------------------
// Continuous convolution (Open3D CConv) + dense layer, fused for MI455X.
//   Phase A (per workgroup of 32 query points): edge scatter into a 32x2048
//           fp32 accumulator tile in LDS (256 KB of the 320 KB CDNA5 WGP LDS).
//   Phase B: 32x2048 @ 2048x64 GEMM via V_WMMA_F32_16X16X4_F32. Each wave32
//           owns one 16-column tile and computes TWO 16x16 output tiles
//           (rows 0-15 and 16-31) that share one B fragment in registers:
//           halves filt L2 traffic and interleaves independent WMMAs to
//           dodge the WMMA->WMMA RAW hazard NOPs.
//   Dense:  feats[N,32] @ W[32,64] + b via the same fp32 WMMA path.
// ---------------------------------------------------------------------------

typedef float v2f __attribute__((ext_vector_type(2)));
typedef float v8f __attribute__((ext_vector_type(8)));

#define CIN       32
#define COUT      64
#define KS        4
#define KVOL      64           // KS^3 spatial kernel positions
#define KC        (KVOL * CIN) // 2048: combined reduction dim (k*CIN + c)
#define QTILE     32           // query rows per workgroup (2 x WMMA M)
#define THREADS   128          // 4 wave32 -> 4 x 16-column output tiles
#define DTILE     16           // rows per block in the dense kernel
#define INV_HALF_EXTENT 25.0f  // 2 / 0.08
#define FOUR_OVER_PI 1.27323954473516f

__device__ __forceinline__ float signf(float v) {
  return (v > 0.0f) ? 1.0f : ((v < 0.0f) ? -1.0f : 0.0f);
}

// qry_idx is globally non-decreasing (chunked row-major nonzero in the
// reference edge builder), so each block can binary-search its edge range.
__device__ __forceinline__ int lower_bound_i32(const int* __restrict__ a,
                                               int n, int key) {
  int lo = 0, hi = n;
  while (lo < hi) {
    int mid = (lo + hi) >> 1;
    if (a[mid] < key) lo = mid + 1; else hi = mid;
  }
  return lo;
}

__global__ __launch_bounds__(THREADS, 1)
void cconv_fused_kernel(const float* __restrict__ feats,     // [N, 32]
                        const float* __restrict__ pos,       // [N, 3]
                        const float* __restrict__ filt,      // [64, 32, 64]
                        const int*   __restrict__ src_idx,   // [E]
                        const int*   __restrict__ qry_idx,   // [E] sorted
                        float*       __restrict__ out_conv,  // [N, 64]
                        int N, int E) {
  extern __shared__ float A_lds[];          // QTILE * KC = 65536 floats (256KB)
  const int tid    = threadIdx.x;
  const int q_base = blockIdx.x * QTILE;

  // ---- zero the accumulator tile -----------------------------------------
  for (int i = tid; i < QTILE * KC; i += THREADS) A_lds[i] = 0.0f;
  __syncthreads();

  // ---- Phase A: edge scatter (LDS float atomics -> ds_add_f32) -----------
  const int e_lo = lower_bound_i32(qry_idx, E, q_base);
  const int e_hi = lower_bound_i32(qry_idx, E, q_base + QTILE);

  for (int e = e_lo + tid; e < e_hi; e += THREADS) {
    const int q  = qry_idx[e];
    const int s  = src_idx[e];
    const int ql = q - q_base;

    const float rx = (pos[3 * s + 0] - pos[3 * q + 0]) * INV_HALF_EXTENT;
    const float ry = (pos[3 * s + 1] - pos[3 * q + 1]) * INV_HALF_EXTENT;
    const float rz = (pos[3 * s + 2] - pos[3 * q + 2]) * INV_HALF_EXTENT;

    // ---- ball_to_cube_volume_preserving (sphere->cylinder->cube) ----
    const float sq   = rx * rx + ry * ry + rz * rz;
    const float nrm3 = sqrtf(sq);
    const bool  pole = (1.25f * rz * rz) > (rx * rx + ry * ry);
    const float s1   = sqrtf(3.0f * nrm3 / (nrm3 + fabsf(rz) + 1e-12f));
    const float nxy  = sqrtf(rx * rx + ry * ry);
    const float s2   = nrm3 / (nxy + 1e-12f);
    float xc = pole ? rx * s1 : rx * s2;
    float yc = pole ? ry * s1 : ry * s2;
    float zc = pole ? signf(rz) * nrm3 : 1.5f * rz;
    if (sq < 1e-12f) { xc = 0.0f; yc = 0.0f; zc = 0.0f; }

    const float sqxy = xc * xc + yc * yc;
    const float nrm2 = sqrtf(sqxy);
    const bool  abr  = fabsf(yc) <= fabsf(xc);
    const float safe_x = (fabsf(xc) < 1e-12f) ? 1.0f : xc;
    const float safe_y = (fabsf(yc) < 1e-12f) ? 1.0f : yc;
    const float tmp_a  = signf(xc) * nrm2;
    const float tmp_b  = signf(yc) * nrm2;
    float xm = abr ? tmp_a : tmp_b * FOUR_OVER_PI * atanf(xc / safe_y);
    float ym = abr ? tmp_a * FOUR_OVER_PI * atanf(yc / safe_x) : tmp_b;
    if (sqxy < 1e-12f) { xm = 0.0f; ym = 0.0f; }
    const float zm = zc;

    // ---- trilinear corners on the 4x4x4 grid ----
    const float half = 0.5f * (KS - 1);                 // 1.5
    const float gx = (xm + 1.0f) * half;
    const float gy = (ym + 1.0f) * half;
    const float gz = (zm + 1.0f) * half;
    const float gx0 = floorf(gx), gy0 = floorf(gy), gz0 = floorf(gz);
    const float fx = gx - gx0, fy = gy - gy0, fz = gz - gz0;
    const int ix[2] = { (int)fminf(fmaxf(gx0,        0.0f), (float)(KS - 1)),
                        (int)fminf(fmaxf(gx0 + 1.0f, 0.0f), (float)(KS - 1)) };
    const int iy[2] = { (int)fminf(fmaxf(gy0,        0.0f), (float)(KS - 1)),
                        (int)fminf(fmaxf(gy0 + 1.0f, 0.0f), (float)(KS - 1)) };
    const int iz[2] = { (int)fminf(fmaxf(gz0,        0.0f), (float)(KS - 1)),
                        (int)fminf(fmaxf(gz0 + 1.0f, 0.0f), (float)(KS - 1)) };
    const float wx[2] = { 1.0f - fx, fx };
    const float wy[2] = { 1.0f - fy, fy };
    const float wz[2] = { 1.0f - fz, fz };

    // ---- gather source feature row (8 x float4, 128B-aligned) ----
    float fr[CIN];
    const float4* f4 = reinterpret_cast<const float4*>(feats + (size_t)s * CIN);
    #pragma unroll
    for (int c4 = 0; c4 < CIN / 4; ++c4) {
      float4 v = f4[c4];
      fr[4 * c4 + 0] = v.x; fr[4 * c4 + 1] = v.y;
      fr[4 * c4 + 2] = v.z; fr[4 * c4 + 3] = v.w;
    }

    // ---- scatter into the 8 corners ----
    #pragma unroll
    for (int zb = 0; zb < 2; ++zb)
      #pragma unroll
      for (int yb = 0; yb < 2; ++yb)
        #pragma unroll
        for (int xb = 0; xb < 2; ++xb) {
          const float w = wz[zb] * wy[yb] * wx[xb];
          const int   k = iz[zb] * (KS * KS) + iy[yb] * KS + ix[xb];
          float* dst = &A_lds[ql * KC + k * CIN];
          #pragma unroll
          for (int c = 0; c < CIN; ++c) atomicAdd(&dst[c], w * fr[c]);
        }
  }
  __syncthreads();

  // ---- Phase B: 32x2048 @ 2048x64 via V_WMMA_F32_16X16X4_F32 -------------
  // Uniform control flow: EXEC must be all-ones for WMMA.
  const int lane = tid & 31;
  const int wv   = tid >> 5;          // wave id -> 16-column output tile
  const int m    = lane & 15;         // A row / C column within tile
  const int hi   = lane >> 4;         // half-wave: K pair 0/1 vs 2/3
  const int kh   = hi * 2;
  const int col  = wv * 16 + m;

  const float* __restrict__ arow0 = A_lds + m * KC;          // rows 0..15
  const float* __restrict__ arow1 = A_lds + (16 + m) * KC;   // rows 16..31
  v8f acc0 = {0.f, 0.f, 0.f, 0.f, 0.f, 0.f, 0.f, 0.f};
  v8f acc1 = {0.f, 0.f, 0.f, 0.f, 0.f, 0.f, 0.f, 0.f};
  for (int kk = 0; kk < KC; kk += 4) {
    // B 4x16 fragment (lanes0-15 K0/K1, lanes16-31 K2/K3; N = lane&15),
    // loaded once and shared by the two row-halves in registers.
    const float* bp = filt + (size_t)(kk + kh) * COUT + col;
    v2f b;
    b.x = bp[0];
    b.y = bp[COUT];
    v2f a0, a1;                     // A 16x4 fragments for the two row tiles
    a0.x = arow0[kk + kh];
    a0.y = arow0[kk + kh + 1];
    a1.x = arow1[kk + kh];
    a1.y = arow1[kk + kh + 1];
    acc0 = __builtin_amdgcn_wmma_f32_16x16x4_f32(
        false, a0, false, b, (short)0, acc0, false, false);
    acc1 = __builtin_amdgcn_wmma_f32_16x16x4_f32(
        false, a1, false, b, (short)0, acc1, false, false);
  }

  // C/D layout: VGPR r holds M=r (lanes 0-15) / M=8+r (lanes 16-31), N=lane&15
  #pragma unroll
  for (int r = 0; r < 8; ++r) {
    const int row0 = q_base + hi * 8 + r;
    if (row0 < N) out_conv[(size_t)row0 * COUT + col] = acc0[r];
  }
  #pragma unroll
  for (int r = 0; r < 8; ++r) {
    const int row1 = q_base + 16 + hi * 8 + r;
    if (row1 < N) out_conv[(size_t)row1 * COUT + col] = acc1[r];
  }
}

__global__ __launch_bounds__(THREADS, 4)
void dense_wmma_kernel(const float* __restrict__ feats,    // [N, 32]
                       const float* __restrict__ dense_w,  // [32, 64]
                       const float* __restrict__ dense_b,  // [64]
                       float*       __restrict__ out,      // [N, 64]
                       int N) {
  const int tid    = threadIdx.x;
  const int q_base = blockIdx.x * DTILE;
  const int lane = tid & 31;
  const int wv   = tid >> 5;
  const int m    = lane & 15;
  const int hi   = lane >> 4;
  const int kh   = hi * 2;
  const int col  = wv * 16 + m;

  const int arow = min(q_base + m, N - 1);   // clamp keeps loads in-bounds
  const float* __restrict__ ap = feats + (size_t)arow * CIN;

  v8f acc = {0.f, 0.f, 0.f, 0.f, 0.f, 0.f, 0.f, 0.f};
  #pragma unroll
  for (int kk = 0; kk < CIN; kk += 4) {
    v2f a;
    a.x = ap[kk + kh];
    a.y = ap[kk + kh + 1];
    const float* bp = dense_w + (size_t)(kk + kh) * COUT + col;
    v2f b;
    b.x = bp[0];
    b.y = bp[COUT];
    acc = __builtin_amdgcn_wmma_f32_16x16x4_f32(
        false, a, false, b, (short)0, acc, false, false);
  }

  const float bias = dense_b[col];
  #pragma unroll
  for (int r = 0; r < 8; ++r) {
    const int row = q_base + hi * 8 + r;
    if (row < N) out[(size_t)row * COUT + col] = acc[r] + bias;
  }
}

extern "C" void kernel_launch(void* const* d_in, const int* in_sizes, int n_in,
                              void* d_out, int out_size, void* d_ws, size_t ws_size,
                              hipStream_t stream) {
  const float* feats   = (const float*)d_in[0];   // [N, 32]
  const float* pos     = (const float*)d_in[1];   // [N, 3]
  const float* filt    = (const float*)d_in[2];   // [4,4,4,32,64]
  const float* dense_w = (const float*)d_in[3];   // [32, 64]
  const float* dense_b = (const float*)d_in[4];   // [64]
  const int*   src_idx = (const int*)d_in[5];     // [E]
  const int*   qry_idx = (const int*)d_in[6];     // [E]

  const int N = in_sizes[0] / CIN;
  const int E = in_sizes[5];

  float* out_conv  = (float*)d_out;                       // first N*64 floats
  float* out_dense = out_conv + (size_t)N * COUT;         // next  N*64 floats

  const int conv_blocks  = (N + QTILE - 1) / QTILE;       // 938 for N=30000
  const int dense_blocks = (N + DTILE - 1) / DTILE;       // 1875 for N=30000
  const size_t shmem = (size_t)QTILE * KC * sizeof(float);// 256 KB of 320 KB WGP LDS

  hipLaunchKernelGGL(cconv_fused_kernel, dim3(conv_blocks), dim3(THREADS), shmem,
                     stream, feats, pos, filt, src_idx, qry_idx, out_conv, N, E);
  hipLaunchKernelGGL(dense_wmma_kernel, dim3(dense_blocks), dim3(THREADS), 0,
                     stream, feats, dense_w, dense_b, out_dense, N);
}